// NPairsLoss_19104014532689
// MI455X (gfx1250) — compile-verified
//
#include <hip/hip_runtime.h>
#include <math.h>

typedef __attribute__((ext_vector_type(2))) float v2f;
typedef __attribute__((ext_vector_type(8))) float v8f;

#define N_ROWS 8192
#define DIM    128
#define NSPLIT 4
#define JTILES_TOTAL (N_ROWS / 16)            /* 512 */
#define JTILES_PER_SPLIT (JTILES_TOTAL / NSPLIT)

#define M_SASC 0.3f
#define M_SADC 0.2f
#define M_DASC 0.4f

// stats layout in workspace (after sq[N_ROWS]):
//  [0]=max_sasc [1]=max_sadc [2]=max_dasc [3]=min_sadc [4]=min_dasc [5]=min_dadc

__global__ void init_stats_kernel(float* __restrict__ stats) {
    int i = blockIdx.x * blockDim.x + threadIdx.x;
    if (i < 3 * N_ROWS)       stats[i] = 0.0f;              // max identities (dist > 0)
    else if (i < 6 * N_ROWS)  stats[i] = __builtin_inff();  // min identities
}

__global__ void sqnorm_kernel(const float* __restrict__ X, float* __restrict__ sq) {
    int wave = (blockIdx.x * blockDim.x + threadIdx.x) >> 5;   // one wave per row
    int lane = threadIdx.x & 31;
    const float4 v = *(const float4*)(X + wave * DIM + lane * 4);
    float s = v.x * v.x + v.y * v.y + v.z * v.z + v.w * v.w;
#pragma unroll
    for (int m = 16; m >= 1; m >>= 1) s += __shfl_xor(s, m, 32);
    if (lane == 0) sq[wave] = s;
}

__global__ void __launch_bounds__(32)
dist_stats_kernel(const float* __restrict__ X, const int* __restrict__ tgt,
                  const float* __restrict__ sq, float* __restrict__ stats) {
    const int lane = threadIdx.x;
    const int hh   = lane >> 4;      // which half of the wave
    const int l    = lane & 15;
    const int rowBase = blockIdx.x * 16;
    const int jt0     = blockIdx.y * JTILES_PER_SPLIT;

    // Resident A panel: 16 rows x 128 K in native WMMA f32 16x16x4 layout.
    // lane l in [0,16): M=l, K pair {4k+0,4k+1}; lanes [16,32): M=l, K pair {4k+2,4k+3}
    v2f a[32];
    const float* arow = X + (size_t)(rowBase + l) * DIM + 2 * hh;
#pragma unroll
    for (int k = 0; k < 32; ++k) a[k] = *(const v2f*)(arow + 4 * k);

    // Per-lane row metadata: this lane's C/D rows are M = 8*hh + v, v=0..7
    int   trow[8]; float sqrow[8];
#pragma unroll
    for (int v = 0; v < 8; ++v) {
        int r    = rowBase + 8 * hh + v;
        trow[v]  = tgt[r >> 2];
        sqrow[v] = sq[r];
    }

    float mx_sasc[8], mx_sadc[8], mx_dasc[8];
    float mn_sadc[8], mn_dasc[8], mn_dadc[8];
#pragma unroll
    for (int v = 0; v < 8; ++v) {
        mx_sasc[v] = mx_sadc[v] = mx_dasc[v] = 0.0f;
        mn_sadc[v] = mn_dasc[v] = mn_dadc[v] = __builtin_inff();
    }

    for (int jt = jt0; jt < jt0 + JTILES_PER_SPLIT; ++jt) {
        const int   col   = jt * 16 + l;          // this lane's N column
        const float* brow = X + (size_t)col * DIM + 2 * hh;
        const int   tcol  = tgt[col >> 2];
        const int   pcol  = col & 3;
        const float sqcol = sq[col];

        v8f acc = {};
#pragma unroll
        for (int g = 0; g < 2; ++g) {
            v2f b[16];
#pragma unroll
            for (int k = 0; k < 16; ++k) b[k] = *(const v2f*)(brow + 4 * (g * 16 + k));
#pragma unroll
            for (int k = 0; k < 16; ++k)
                acc = __builtin_amdgcn_wmma_f32_16x16x4_f32(
                          false, a[g * 16 + k], false, b[k],
                          (short)0, acc, false, false);
        }

#pragma unroll
        for (int v = 0; v < 8; ++v) {
            float dd   = sqrow[v] + sqcol - 2.0f * acc[v];
            dd         = fmaxf(dd, 1e-12f);
            float dist = __builtin_sqrtf(dd);
            bool sc = (trow[v] == tcol);
            bool sp = ((v & 3) == pcol);          // part = row & 3; rowBase,8*hh are mult of 4
            mx_sasc[v] = fmaxf(mx_sasc[v], ( sc &&  sp) ? dist : 0.0f);
            mx_sadc[v] = fmaxf(mx_sadc[v], (!sc &&  sp) ? dist : 0.0f);
            mx_dasc[v] = fmaxf(mx_dasc[v], ( sc && !sp) ? dist : 0.0f);
            mn_sadc[v] = fminf(mn_sadc[v], (!sc &&  sp) ? dist : __builtin_inff());
            mn_dasc[v] = fminf(mn_dasc[v], ( sc && !sp) ? dist : __builtin_inff());
            mn_dadc[v] = fminf(mn_dadc[v], (!sc && !sp) ? dist : __builtin_inff());
        }
    }

    // Reduce across the 16 lanes that share rows M = 8*hh + v
#pragma unroll
    for (int v = 0; v < 8; ++v) {
#pragma unroll
        for (int m = 1; m < 16; m <<= 1) {
            mx_sasc[v] = fmaxf(mx_sasc[v], __shfl_xor(mx_sasc[v], m, 32));
            mx_sadc[v] = fmaxf(mx_sadc[v], __shfl_xor(mx_sadc[v], m, 32));
            mx_dasc[v] = fmaxf(mx_dasc[v], __shfl_xor(mx_dasc[v], m, 32));
            mn_sadc[v] = fminf(mn_sadc[v], __shfl_xor(mn_sadc[v], m, 32));
            mn_dasc[v] = fminf(mn_dasc[v], __shfl_xor(mn_dasc[v], m, 32));
            mn_dadc[v] = fminf(mn_dadc[v], __shfl_xor(mn_dadc[v], m, 32));
        }
    }

    if (l == 0) {
        int rb = rowBase + 8 * hh;
#pragma unroll
        for (int v = 0; v < 8; ++v) {
            // nonnegative floats: IEEE order == signed-int order of bit patterns
            atomicMax((int*)(stats + 0 * N_ROWS + rb + v), __float_as_int(mx_sasc[v]));
            atomicMax((int*)(stats + 1 * N_ROWS + rb + v), __float_as_int(mx_sadc[v]));
            atomicMax((int*)(stats + 2 * N_ROWS + rb + v), __float_as_int(mx_dasc[v]));
            atomicMin((int*)(stats + 3 * N_ROWS + rb + v), __float_as_int(mn_sadc[v]));
            atomicMin((int*)(stats + 4 * N_ROWS + rb + v), __float_as_int(mn_dasc[v]));
            atomicMin((int*)(stats + 5 * N_ROWS + rb + v), __float_as_int(mn_dadc[v]));
        }
    }
}

__global__ void loss_kernel(const float* __restrict__ stats, float* __restrict__ out) {
    __shared__ float red[256];
    int tid = threadIdx.x;
    float s = 0.0f;
    for (int r = tid; r < N_ROWS; r += 256) {
        float ap_sasc = stats[0 * N_ROWS + r];
        float ap_sadc = stats[1 * N_ROWS + r];
        float ap_dasc = stats[2 * N_ROWS + r];
        float mn_sadc = stats[3 * N_ROWS + r];
        float mn_dasc = stats[4 * N_ROWS + r];
        float mn_dadc = stats[5 * N_ROWS + r];
        float an_sasc = fminf(mn_sadc, fminf(mn_dasc, mn_dadc));
        s += fmaxf(0.0f, M_SASC - (an_sasc - ap_sasc));
        s += fmaxf(0.0f, M_SADC - (mn_dadc - ap_sadc));
        s += fmaxf(0.0f, M_DASC - (mn_dadc - ap_dasc));
    }
    red[tid] = s;
    __syncthreads();
    for (int st = 128; st > 0; st >>= 1) {
        if (tid < st) red[tid] += red[tid + st];
        __syncthreads();
    }
    if (tid == 0) out[0] = red[0] / (float)N_ROWS;
}

extern "C" void kernel_launch(void* const* d_in, const int* in_sizes, int n_in,
                              void* d_out, int out_size, void* d_ws, size_t ws_size,
                              hipStream_t stream) {
    (void)in_sizes; (void)n_in; (void)out_size; (void)ws_size;
    const float* X   = (const float*)d_in[0];
    const int*   tgt = (const int*)d_in[1];
    float* sq    = (float*)d_ws;
    float* stats = sq + N_ROWS;

    init_stats_kernel<<<(6 * N_ROWS + 255) / 256, 256, 0, stream>>>(stats);
    sqnorm_kernel<<<(N_ROWS * 32) / 256, 256, 0, stream>>>(X, sq);
    dim3 grid(N_ROWS / 16, NSPLIT);
    dist_stats_kernel<<<grid, 32, 0, stream>>>(X, tgt, sq, stats);
    loss_kernel<<<1, 256, 0, stream>>>(stats, (float*)d_out);
}